// LineSpectralPairsToLinearPredictiveCoefficients_50216757625148
// MI455X (gfx1250) — compile-verified
//
#include <hip/hip_runtime.h>

typedef __attribute__((ext_vector_type(2))) float v2f;
typedef __attribute__((ext_vector_type(8))) float v8f;

#define KSTRIDE 29   // LDS row stride in dwords; coprime with 64 banks -> conflict-free

// ---- compile-time constant combine matrix, padded to 32 rows x 28 cols ----
// rows 24..31 are zero so B-fragment loads need no predication (zero rows
// contribute nothing through the WMMA).
constexpr double binom_ct(int n, int r) {
  if (r < 0 || r > n) return 0.0;
  double res = 1.0;
  for (int i = 0; i < r; ++i) res = res * (double)(n - i) / (double)(i + 1);
  return res;
}
// Coefficient of x^(24-m) contributed by y^k after x^12*(x+1/x)^k expansion
constexpr double Cm_ct(int k, int m) {
  const int s = k + m - 12;
  if (s < 0 || (s & 1)) return 0.0;
  const int i = s / 2;
  if (i > k) return 0.0;
  return binom_ct(k, i);
}
struct BmatT { float v[32][28]; };
constexpr BmatT makeB() {
  BmatT b{};
  for (int t = 0; t < 32; ++t)
    for (int c = 0; c < 28; ++c) b.v[t][c] = 0.0f;
  for (int t = 0; t < 24; ++t) {
    for (int k = 0; k <= 12; ++k) {
      // a[t] = sum_k rP_k * 0.5*(C(k,t+1)-C(k,t)) + sum_k rQ_k * 0.5*(C(k,t)+C(k,t+1))
      b.v[t][k]      = (float)(0.5 * (Cm_ct(k, t + 1) - Cm_ct(k, t)));
      b.v[t][13 + k] = (float)(0.5 * (Cm_ct(k, t) + Cm_ct(k, t + 1)));
    }
  }
  return b;
}
__device__ __constant__ BmatT gB = makeB();

__global__ __launch_bounds__(256) void lsp2lpc_wmma(const float* __restrict__ w,
                                                    float* __restrict__ out,
                                                    int nItems) {
  // 8 waves * 32 items * 29 dwords = 29,696 B
  __shared__ float lds[8][32][KSTRIDE];

  const int tid  = threadIdx.x;
  const int wave = tid >> 5;
  const int lane = tid & 31;
  const int nsub = lane & 15;   // N (or M) within a 16-wide tile
  const int hi   = lane >> 4;   // half-wave select (K pair 0/1 vs 2/3)

  const long long blkBase = (long long)blockIdx.x * 256;
  const long long item    = blkBase + tid;
  const long long itemL   = (item < (long long)nItems) ? item : (long long)(nItems - 1);

  // ---------- phase 1: per-item symmetric-function recurrence (VALU) ----------
  const float* wp = w + itemL * 25;
  const float K = wp[0];

  // g*[d] = descending coeff of prod_j (y - 2*cos(theta_j)); monic degree 12
  float gP[13], gQ[13];
  gP[0] = 1.f; gQ[0] = 1.f;
#pragma unroll
  for (int i = 1; i < 13; ++i) { gP[i] = 0.f; gQ[i] = 0.f; }

#pragma unroll
  for (int j = 0; j < 12; ++j) {
    // p roots: odd-index angles -> wp[2j+2]; q roots: even-index -> wp[2j+1]
    const float ap = -2.f * __cosf(wp[2 * j + 2]);
    const float aq = -2.f * __cosf(wp[2 * j + 1]);
#pragma unroll
    for (int d = j + 1; d >= 1; --d) {
      gP[d] = __builtin_fmaf(ap, gP[d - 1], gP[d]);
      gQ[d] = __builtin_fmaf(aq, gQ[d - 1], gQ[d]);
    }
  }

  // row vector v[0..27] = [rP_0..rP_12, rQ_0..rQ_12, 0, 0]; r_k = g[12-k]
  float* myrow = &lds[wave][lane][0];
#pragma unroll
  for (int kk = 0; kk < 13; ++kk) myrow[kk]      = gP[12 - kk];
#pragma unroll
  for (int kk = 0; kk < 13; ++kk) myrow[13 + kk] = gQ[12 - kk];
  myrow[26] = 0.f; myrow[27] = 0.f;

  if (item < (long long)nItems) out[item * 25] = K;  // gain passthrough

  __syncthreads();

  // ---------- phase 2: (32 items x 28) * B^T(28 x 32) via V_WMMA_F32_16X16X4_F32 ----------
  // B fragments (constant matrix): lane = N within tile, v0/v1 = K pair per half-wave.
  // Unconditional loads: table rows 24..31 are zero-padded.
  v2f bfrag[2][7];
#pragma unroll
  for (int nt = 0; nt < 2; ++nt) {
    const int t = nt * 16 + nsub;
#pragma unroll
    for (int ks = 0; ks < 7; ++ks) {
      const int c = ks * 4 + hi * 2;
      bfrag[nt][ks].x = gB.v[t][c];
      bfrag[nt][ks].y = gB.v[t][c + 1];
    }
  }

  v8f acc[2][2] = {};

#pragma unroll
  for (int mt = 0; mt < 2; ++mt) {
    const float* arow = &lds[wave][mt * 16 + nsub][0];  // A: lane = M%16
#pragma unroll
    for (int ks = 0; ks < 7; ++ks) {
      const int c = ks * 4 + hi * 2;
      v2f af;
      af.x = arow[c];
      af.y = arow[c + 1];
#pragma unroll
      for (int nt = 0; nt < 2; ++nt) {
        // 8 args: (neg_a, A, neg_b, B, c_mod, C, reuse_a, reuse_b)
        acc[mt][nt] = __builtin_amdgcn_wmma_f32_16x16x4_f32(
            false, af, false, bfrag[nt][ks], (short)0, acc[mt][nt], false, false);
      }
    }
  }

  // ---------- store: D tile layout (vgpr g -> row g / g+8 per half-wave) ----------
#pragma unroll
  for (int mt = 0; mt < 2; ++mt) {
#pragma unroll
    for (int nt = 0; nt < 2; ++nt) {
      const int t = nt * 16 + nsub;  // output coefficient index (0..23 valid)
#pragma unroll
      for (int g = 0; g < 8; ++g) {
        const long long it = blkBase + (long long)wave * 32 + mt * 16 + g + hi * 8;
        if (t < 24 && it < (long long)nItems) {
          out[it * 25 + 1 + t] = acc[mt][nt][g];
        }
      }
    }
  }
}

extern "C" void kernel_launch(void* const* d_in, const int* in_sizes, int n_in,
                              void* d_out, int out_size, void* d_ws, size_t ws_size,
                              hipStream_t stream) {
  const float* w = (const float*)d_in[0];
  float* out = (float*)d_out;
  const int nItems = in_sizes[0] / 25;            // (512*2048) items of 25 floats
  const int blocks = (nItems + 255) / 256;
  lsp2lpc_wmma<<<blocks, 256, 0, stream>>>(w, out, nItems);
}